// AGNNConv_17712445129506
// MI455X (gfx1250) — compile-verified
//
#include <hip/hip_runtime.h>
#include <hip/hip_bf16.h>

#define D_FEAT 64
#define EPS 1e-7f

typedef __attribute__((ext_vector_type(2))) float v2f;
typedef __attribute__((ext_vector_type(8))) float v8f;

// ---------------------------------------------------------------------------
// Kernel 1: per-node init. One wave (32 lanes) per node:
//  - compute nrm[i] = ||x[i]|| via float2 loads + shfl_xor reduction
//  - zero out[i,:]
//  - init segMax (ordered-uint encoding, 0 == -lowest) and segSum
// ---------------------------------------------------------------------------
__global__ void agnn_init_norm(const float* __restrict__ x,
                               float* __restrict__ nrm,
                               unsigned* __restrict__ segMaxU,
                               float* __restrict__ segSum,
                               float* __restrict__ out,
                               int nNodes) {
    const int lane = threadIdx.x & 31;
    const int node = blockIdx.x * (blockDim.x >> 5) + (threadIdx.x >> 5);
    if (node >= nNodes) return;

    const float2 v = ((const float2*)(x + (size_t)node * D_FEAT))[lane];
    float s = v.x * v.x + v.y * v.y;
#pragma unroll
    for (int off = 16; off > 0; off >>= 1)
        s += __shfl_xor(s, off, 32);

    float2 z; z.x = 0.f; z.y = 0.f;
    ((float2*)(out + (size_t)node * D_FEAT))[lane] = z;

    if (lane == 0) {
        nrm[node] = sqrtf(s);
        segMaxU[node] = 0u;   // minimum key in ordered-uint space
        segSum[node] = 0.f;
    }
}

// ---------------------------------------------------------------------------
// Kernel 2: edge similarity via WMMA. One wave handles 16 edges.
// A fragment (16x4 f32): A[m][k] = x[row[eBase+m]][4*kc+k]
//   lane layout: lanes 0-15 -> M=lane, K={0,1}; lanes 16-31 -> M=lane-16, K={2,3}
// B fragment (4x16 f32): B[k][n] = x[col[eBase+n]][4*kc+k]
//   lane layout: lanes 0-15 -> N=lane, K={0,1}; lanes 16-31 -> N=lane-16, K={2,3}
// => one global_load_b64 per lane per fragment per K-chunk; 16 chained WMMAs
// accumulate K=64. diag(C)[m] = <x[row[e]], x[col[e]]>.
// ---------------------------------------------------------------------------
__global__ void agnn_edge_sim(const float* __restrict__ x,
                              const long long* __restrict__ row,
                              const long long* __restrict__ col,
                              const float* __restrict__ nrm,
                              const float* __restrict__ betaPtr,
                              float* __restrict__ simBuf,
                              unsigned* __restrict__ segMaxU,
                              int nEdges) {
    const int lane  = threadIdx.x & 31;
    const int wave  = threadIdx.x >> 5;
    const int eBase = (blockIdx.x * (blockDim.x >> 5) + wave) << 4;  // 16 edges/wave

    // fragment-gather edge for this lane (clamped so EXEC stays all-ones)
    int eFrag = eBase + (lane & 15);
    if (eFrag >= nEdges) eFrag = nEdges - 1;
    const int r = (int)row[eFrag];
    const int c = (int)col[eFrag];
    const int kOff = (lane < 16) ? 0 : 2;
    const float* pa = x + (size_t)r * D_FEAT + kOff;
    const float* pb = x + (size_t)c * D_FEAT + kOff;

    v8f acc = {0.f, 0.f, 0.f, 0.f, 0.f, 0.f, 0.f, 0.f};
#pragma unroll
    for (int kc = 0; kc < 16; ++kc) {
        const v2f a = *(const v2f*)(pa + kc * 4);
        const v2f b = *(const v2f*)(pb + kc * 4);
        acc = __builtin_amdgcn_wmma_f32_16x16x4_f32(
            /*neg_a=*/false, a, /*neg_b=*/false, b,
            /*c_mod=*/(short)0, acc, /*reuse_a=*/false, /*reuse_b=*/false);
    }

    // diag extraction: m<8 -> lane=m, vgpr=m ; m>=8 -> lane=m+8, vgpr=m-8
    const bool act = (lane < 8) || (lane >= 16 && lane < 24);
    const int dIdx = (lane < 8) ? lane : (lane - 16) & 7;
    const float lo01 = (dIdx & 1) ? acc[1] : acc[0];
    const float lo23 = (dIdx & 1) ? acc[3] : acc[2];
    const float lo45 = (dIdx & 1) ? acc[5] : acc[4];
    const float lo67 = (dIdx & 1) ? acc[7] : acc[6];
    const float q0   = (dIdx & 2) ? lo23 : lo01;
    const float q1   = (dIdx & 2) ? lo67 : lo45;
    const float inner = (dIdx & 4) ? q1 : q0;

    if (act) {
        const int eD = eBase + ((lane < 8) ? lane : (lane - 8));
        if (eD < nEdges) {
            const int rr = (int)row[eD];
            const int cc = (int)col[eD];
            const float denom = nrm[rr] * nrm[cc] + EPS;
            const float sim = betaPtr[0] * (inner / denom);
            simBuf[eD] = sim;
            // ordered-uint encoding for float atomic max
            const unsigned u = __float_as_uint(sim);
            const unsigned key = (u & 0x80000000u) ? ~u : (u | 0x80000000u);
            atomicMax(&segMaxU[rr], key);
        }
    }
}

// ---------------------------------------------------------------------------
// Kernel 3: w = exp(sim - segMax[row]); segSum[row] += w (native f32 atomic)
// ---------------------------------------------------------------------------
__global__ void agnn_edge_exp(const long long* __restrict__ row,
                              float* __restrict__ simBuf,
                              const unsigned* __restrict__ segMaxU,
                              float* __restrict__ segSum,
                              int nEdges) {
    const int e = blockIdx.x * blockDim.x + threadIdx.x;
    if (e >= nEdges) return;
    const int r = (int)row[e];
    const unsigned key = segMaxU[r];
    const unsigned u = (key & 0x80000000u) ? (key & 0x7FFFFFFFu) : ~key;
    const float m = __uint_as_float(u);
    const float w = __expf(simBuf[e] - m);
    simBuf[e] = w;
    unsafeAtomicAdd(&segSum[r], w);
}

// ---------------------------------------------------------------------------
// Kernel 4: scatter. One wave per edge: P = w/segSum[row];
// out[col, :] += x[row, :] * P  (float2 load, 2 native f32 atomics per lane)
// ---------------------------------------------------------------------------
__global__ void agnn_edge_scatter(const float* __restrict__ x,
                                  const long long* __restrict__ row,
                                  const long long* __restrict__ col,
                                  const float* __restrict__ wBuf,
                                  const float* __restrict__ segSum,
                                  float* __restrict__ out,
                                  int nEdges) {
    const int lane = threadIdx.x & 31;
    const int e = blockIdx.x * (blockDim.x >> 5) + (threadIdx.x >> 5);
    if (e >= nEdges) return;
    const int r = (int)row[e];
    const int c = (int)col[e];
    const float P = wBuf[e] / segSum[r];
    const float2 v = ((const float2*)(x + (size_t)r * D_FEAT))[lane];
    float* po = out + (size_t)c * D_FEAT + 2 * lane;
    unsafeAtomicAdd(po,     v.x * P);
    unsafeAtomicAdd(po + 1, v.y * P);
}

// ---------------------------------------------------------------------------
extern "C" void kernel_launch(void* const* d_in, const int* in_sizes, int n_in,
                              void* d_out, int out_size, void* d_ws, size_t ws_size,
                              hipStream_t stream) {
    const float*     x    = (const float*)d_in[0];
    const long long* row  = (const long long*)d_in[1];
    const long long* col  = (const long long*)d_in[2];
    const float*     beta = (const float*)d_in[3];
    float* out = (float*)d_out;

    const int nEdges = in_sizes[1];
    const int nNodes = in_sizes[0] / D_FEAT;

    // workspace layout
    char* ws = (char*)d_ws;
    float*    simBuf  = (float*)ws;                                   // nEdges f32
    unsigned* segMaxU = (unsigned*)(ws + (size_t)nEdges * 4);         // nNodes u32
    float*    segSum  = (float*)(ws + (size_t)nEdges * 4
                                    + (size_t)nNodes * 4);            // nNodes f32
    float*    nrm     = (float*)(ws + (size_t)nEdges * 4
                                    + (size_t)nNodes * 8);            // nNodes f32

    const int BLK = 256;                    // 8 waves/block
    const int wavesPerBlock = BLK >> 5;

    // 1) norms + zero out + init segment state (one wave per node)
    {
        const int grid = (nNodes + wavesPerBlock - 1) / wavesPerBlock;
        agnn_init_norm<<<grid, BLK, 0, stream>>>(x, nrm, segMaxU, segSum, out, nNodes);
    }
    // 2) WMMA similarity + segment max (16 edges per wave -> 128 per block)
    {
        const int edgesPerBlock = wavesPerBlock * 16;
        const int grid = (nEdges + edgesPerBlock - 1) / edgesPerBlock;
        agnn_edge_sim<<<grid, BLK, 0, stream>>>(x, row, col, nrm, beta,
                                                simBuf, segMaxU, nEdges);
    }
    // 3) exp + segment sum (thread per edge)
    {
        const int grid = (nEdges + BLK - 1) / BLK;
        agnn_edge_exp<<<grid, BLK, 0, stream>>>(row, simBuf, segMaxU, segSum, nEdges);
    }
    // 4) normalize + scatter-add (wave per edge)
    {
        const int grid = (nEdges + wavesPerBlock - 1) / wavesPerBlock;
        agnn_edge_scatter<<<grid, BLK, 0, stream>>>(x, row, col, simBuf, segSum,
                                                    out, nEdges);
    }
}